// TSPSolver_25305947308612
// MI455X (gfx1250) — compile-verified
//
#include <hip/hip_runtime.h>
#include <math.h>

// ---------------------------------------------------------------------------
// TSP spiking-KWTA solver for MI455X (gfx1250).
//
// The computation is one serial dependency chain (t -> block -> 2 KWTA
// substeps = 256K serial iterations), so we run ONE wave32 and minimize the
// per-iteration critical path:
//   * W (256x256 f32 = 256KB) lives in the WGP's 320KB LDS, staged once by the
//     Tensor Data Mover with pad_enable (1 DWORD pad per 256 DWORDs) -> row
//     stride 257 -> the one-hot "x @ W.T" gather W[j][idx] is bank-conflict
//     free ((257j+idx) mod 64 = (j+idx) mod 64).
//   * Per-(t,block) noise slices (512 f32, contiguous) prefetched 16 deep via
//     a TDM ring; s_wait_tensorcnt(DEPTH-1) proves the consumed slot complete
//     (TENSORcnt completes in order per wave).
//   * data.sum(0) kept as per-lane REGISTER spike-count histograms (8 cities
//     per lane); winners are wave-uniform so +-1 updates use unrolled
//     compares. Small-integer float adds => bit-exact vs the reference.
//   * argmax with jnp first-index tie-break: per-lane scan + 5-step
//     __shfl_xor butterfly on (value, index).
// WMMA is deliberately not used: the A operand is one-hot, so WMMA would add
// >100 matrix ops per substep to a latency-bound chain a single ds_load solves.
// ---------------------------------------------------------------------------

#define NCITY 256
#define TSTEPS 500
#define DEPTH 16             // noise prefetch depth (power of 2)

#define W_BYTES    (256 * 257 * 4)           // padded W: stride 257 floats
#define RING_OFF   W_BYTES
#define RING_BYTES (DEPTH * 512 * 4)
#define WIN_OFF    (RING_OFF + RING_BYTES)
#define SMEM_BYTES (WIN_OFF + 2 * 256 * 4)   // win0, win1

typedef __attribute__((ext_vector_type(4))) unsigned int u32x4_t;
typedef __attribute__((ext_vector_type(4))) int          i32x4_t;
typedef __attribute__((ext_vector_type(8))) int          i32x8_t;

// Tensor DMA descriptor build + issue (2D tile, optional LDS pad-on-store).
// D# layout per CDNA5 ISA ch.10.8 (group0: count/lds/global/type,
// group1: flags/dims/tiles/stride0).
__device__ __forceinline__ void tdm_load_2d(unsigned lds_addr,
                                            unsigned long long gaddr,
                                            unsigned flags,     // data_size/pad bits
                                            unsigned dim0, unsigned dim1,
                                            unsigned tile0, unsigned tile1,
                                            unsigned stride0) {
  u32x4_t g0;
  g0.x = 1u;                                          // count=1, user D#
  g0.y = lds_addr;                                    // lds_addr [63:32]
  g0.z = (unsigned)(gaddr & 0xFFFFFFFFull);           // global_addr[31:0]
  g0.w = (unsigned)((gaddr >> 32) & 0x1FFFFFFull)     // global_addr[56:32]
       | (2u << 30);                                  // type=2 ("image")
  i32x8_t g1;
  g1.s0 = (int)flags;                                 // mask=0|data_size|pad ctl
  g1.s1 = (int)((dim0 & 0xFFFFu) << 16);              // abar=0 | tensor_dim0 lo
  g1.s2 = (int)((dim0 >> 16) | ((dim1 & 0xFFFFu) << 16));
  g1.s3 = (int)((dim1 >> 16) | (tile0 << 16));        // tile_dim0
  g1.s4 = (int)(tile1 & 0xFFFFu);                     // tile_dim1 | tile_dim2=0
  g1.s5 = (int)stride0;                               // tensor_dim0_stride lo32
  g1.s6 = 0;
  g1.s7 = 0;
  i32x4_t z4 = {0, 0, 0, 0};
#if defined(__clang_major__) && (__clang_major__ >= 23)
  i32x8_t z8 = {0, 0, 0, 0, 0, 0, 0, 0};
  __builtin_amdgcn_tensor_load_to_lds(g0, g1, z4, z4, z8, 0);
#else
  __builtin_amdgcn_tensor_load_to_lds(g0, g1, z4, z4, 0);
#endif
}

// Wave32 argmax with jnp.argmax tie-break (lowest index wins ties).
__device__ __forceinline__ void argmax32(float& v, int& j) {
#pragma unroll
  for (int off = 16; off > 0; off >>= 1) {
    float v2 = __shfl_xor(v, off, 32);
    int   j2 = __shfl_xor(j, off, 32);
    if (v2 > v || (v2 == v && j2 < j)) { v = v2; j = j2; }
  }
}

__global__ __launch_bounds__(32) void tsp_kwta_solver_gfx1250(
    const float* __restrict__ gW,        // (256,256) path_data
    const float* __restrict__ gNoise,    // (500,256,1,256,2)
    float* __restrict__ out)             // [500 dists][500*257 paths] as f32
{
  extern __shared__ char smem[];
  float* Wl   = (float*)smem;                         // padded, stride 257
  float* ring = (float*)(smem + RING_OFF);            // DEPTH x 512 floats
  int*   win0 = (int*)(smem + WIN_OFF);               // per-block winner s=0
  int*   win1 = win0 + 256;                           // per-block winner s=1

  const int lane = threadIdx.x;

  // Register-resident spike-count histograms: lane owns cities r*32+lane.
  float c0[8], c1[8];
#pragma unroll
  for (int r = 0; r < 8; r++) {
    c0[r] = 0.f; c1[r] = 0.f;
    int j = r * 32 + lane;
    win0[j] = -1; win1[j] = -1;
  }

  // flags: data_size=4B (2<<16); W adds pad_enable, pad_interval=256 DWORDs
  // (enc 7), pad_amount=1 DWORD (enc 0) -> LDS row stride 257 floats.
  const unsigned FLAGS_W = (2u << 16) | (1u << 20) | (7u << 22);
  const unsigned FLAGS_N = (2u << 16);

  // One TDM op stages all of W (256KB) into LDS with padding.
  tdm_load_2d((unsigned)(size_t)Wl, (unsigned long long)(size_t)gW,
              FLAGS_W, 256, 256, 256, 256, 256);
  // Prime the noise prefetch ring.
  for (int p = 0; p < DEPTH; p++)
    tdm_load_2d((unsigned)(size_t)(ring + (size_t)p * 512),
                (unsigned long long)(size_t)(gNoise + (size_t)p * 512),
                FLAGS_N, 512, 1, 512, 1, 512);

  for (int t = 0; t < TSTEPS; t++) {
    const float tdn = 0.4f * expf(-2e-5f * (float)t);   // TEMP * decay
    int pw0 = 0, pw1 = 0;      // inp_spike = start_spike (index 0, fired)
    int allf = 1;              // "valid": every block fired at substep 1

    for (int i = 0; i < NCITY; i++) {
      const int bi = t * NCITY + i;
      // Oldest outstanding TDM (this block's noise + W at startup) complete.
      __builtin_amdgcn_s_wait_tensorcnt((short)(DEPTH - 1));
      const float2* nz = (const float2*)(ring + (size_t)(bi & (DEPTH - 1)) * 512);

      // res = data.sum(0) - data[i] (+ start_spike unless i==n-1):
      // remove block i's old contribution (permanent; new winner re-added
      // below == S - old + new). ow0/ow1 are wave-uniform LDS reads.
      const int ow0 = win0[i];
      const int ow1 = win1[i];
#pragma unroll
      for (int r = 0; r < 8; r++) {
        int j = r * 32 + lane;
        if (j == ow0) c0[r] -= 1.f;
        if (j == ow1) c1[r] -= 1.f;
      }
      const float sf = (i != NCITY - 1) ? 1.f : 0.f;   // start_spike indicator

      // ---- KWTA substep 0: v = cur0 = W[:,x0] + bias0 ----
      const int   g0idx = (pw0 >= 0) ? pw0 : 0;
      const float g0s   = (pw0 >= 0) ? 1.f : 0.f;
      float v0[8], nse1[8];
      float best = 0.f; int bidx = 0;
#pragma unroll
      for (int r = 0; r < 8; r++) {
        int j = r * 32 + lane;
        float2 nv = nz[j];                       // ds_load_b64, conflict-free
        nse1[r] = nv.y;
        float bias = -1.5f * (c0[r] + ((j == 0) ? sf : 0.f)) + tdn * nv.x;
        float v = Wl[j * 257 + g0idx] * g0s + bias;   // stride-257: no conflicts
        v0[r] = v;
        if (r == 0) { best = v; bidx = j; }
        else if (v > best) { best = v; bidx = j; }    // strict > keeps lowest j
      }
      argmax32(best, bidx);
      const int w0n = (best >= 0.1f) ? bidx : -1;     // spike0 index (-1: none)
#pragma unroll
      for (int r = 0; r < 8; r++) {                   // v *= (1 - spike)
        int j = r * 32 + lane;
        if (j == w0n) v0[r] = 0.f;
      }

      // ---- KWTA substep 1: v = 0.9*v + (W[:,x1] + 0.5*spike0 + bias1) ----
      const int   g1idx = (pw1 >= 0) ? pw1 : 0;
      const float g1s   = (pw1 >= 0) ? 1.f : 0.f;
      float best1 = 0.f; int bidx1 = 0;
#pragma unroll
      for (int r = 0; r < 8; r++) {
        int j = r * 32 + lane;
        float bias = -1.5f * (c1[r] + ((j == 0) ? sf : 0.f)) + tdn * nse1[r];
        float cur = Wl[j * 257 + g1idx] * g1s
                  + ((j == w0n) ? 0.5f : 0.f) + bias;
        float v = 0.9f * v0[r] + cur;
        if (r == 0) { best1 = v; bidx1 = j; }
        else if (v > best1) { best1 = v; bidx1 = j; }
      }
      argmax32(best1, bidx1);
      const int w1n = (best1 >= 0.1f) ? bidx1 : -1;

      // data[i] = out: add new contribution, record winners.
#pragma unroll
      for (int r = 0; r < 8; r++) {
        int j = r * 32 + lane;
        if (j == w0n) c0[r] += 1.f;
        if (j == w1n) c1[r] += 1.f;
      }
      if (lane == 0) { win0[i] = w0n; win1[i] = w1n; }
      pw0 = w0n; pw1 = w1n;                 // carry out -> next block's input
      allf &= (w1n >= 0) ? 1 : 0;

      // Refill the slot we just consumed (DEPTH blocks ahead).
      if (bi + DEPTH < TSTEPS * NCITY)
        tdm_load_2d((unsigned)(size_t)(ring + (size_t)(bi & (DEPTH - 1)) * 512),
                    (unsigned long long)(size_t)(gNoise + (size_t)(bi + DEPTH) * 512),
                    FLAGS_N, 512, 1, 512, 1, 512);
    }

    // Per-t outputs. path[0]=argmax(start row)=0; path[k]=win1[k-1].
    float dist = -1.f;
    if (allf) {
      int p1 = win1[0], p2 = win1[1], pl = win1[255];
      dist = (Wl[p1] + Wl[p2]) + Wl[pl];    // W[0,p1]+W[0,p2]+W[0,p_last]
    }
    if (lane == 0) out[t] = dist;
#pragma unroll
    for (int c = 0; c < 9; c++) {
      int k = c * 32 + lane;
      if (k < 257) {
        float pv = allf ? (float)((k == 0) ? 0 : win1[k - 1]) : -1.f;
        out[TSTEPS + (size_t)t * 257 + k] = pv;
      }
    }
  }
}

extern "C" void kernel_launch(void* const* d_in, const int* in_sizes, int n_in,
                              void* d_out, int out_size, void* d_ws, size_t ws_size,
                              hipStream_t stream) {
  (void)in_sizes; (void)n_in; (void)out_size; (void)d_ws; (void)ws_size;
  const float* W     = (const float*)d_in[0];   // path_data (256,256)
  // d_in[1] = start_spike: structurally one-hot at city 0 (both substeps);
  // encoded directly in the kernel (pw0=pw1=0, start indicator at j==0).
  const float* noise = (const float*)d_in[2];   // (500,256,1,256,2)
  float* out = (float*)d_out;

  hipFuncSetAttribute(reinterpret_cast<const void*>(&tsp_kwta_solver_gfx1250),
                      hipFuncAttributeMaxDynamicSharedMemorySize,
                      (int)SMEM_BYTES);
  // Single wave32 workgroup: the whole problem is one dependency chain.
  tsp_kwta_solver_gfx1250<<<1, 32, SMEM_BYTES, stream>>>(W, noise, out);
}